// GNN_node_53558242181886
// MI455X (gfx1250) — compile-verified
//
#include <hip/hip_runtime.h>
#include <hip/hip_bf16.h>

typedef __attribute__((ext_vector_type(2))) float v2f;
typedef __attribute__((ext_vector_type(8))) float v8f;

#define DD 64
#define DD2 128
#define NLAYERS 4
#define NAF 9
#define NBF 3
#define AVOC 64
#define BVOC 8

// ---------------------------------------------------------------------------
// Atom encoder: h[n][d] = sum_f atom_emb[f][x[n][f]][d]
// ---------------------------------------------------------------------------
__global__ __launch_bounds__(256) void atom_enc_kernel(
    const int* __restrict__ x, const float* __restrict__ atom_emb,
    float* __restrict__ h, int n) {
  int node = blockIdx.x * 4 + (threadIdx.x >> 6);
  int d = threadIdx.x & 63;
  if (node >= n) return;
  float v = 0.0f;
#pragma unroll
  for (int f = 0; f < NAF; ++f) {
    int idx = x[node * NAF + f];
    v += atom_emb[((size_t)f * AVOC + idx) * DD + d];
  }
  h[(size_t)node * DD + d] = v;
}

// ---------------------------------------------------------------------------
// agg = (1 + eps[l]) * h
// ---------------------------------------------------------------------------
__global__ __launch_bounds__(256) void gin_init_kernel(
    const float* __restrict__ h, const float* __restrict__ eps, int l,
    float* __restrict__ agg, long long total) {
  long long i = (long long)blockIdx.x * blockDim.x + threadIdx.x;
  if (i < total) agg[i] = (1.0f + eps[l]) * h[i];
}

// ---------------------------------------------------------------------------
// Edge kernel: msg = relu(h[src] + bond_embed(edge_attr)); atomicAdd into agg[dst]
// 64 lanes per edge (one per channel), 4 edges per 256-thread block.
// ---------------------------------------------------------------------------
__global__ __launch_bounds__(256) void msg_scatter_kernel(
    const float* __restrict__ h, const int* __restrict__ edge_index,
    const int* __restrict__ edge_attr, const float* __restrict__ bond_emb_l,
    float* __restrict__ agg, int n_edges) {
  int e = blockIdx.x * 4 + (threadIdx.x >> 6);
  int d = threadIdx.x & 63;
  if (e >= n_edges) return;
  int src = edge_index[e];
  int dst = edge_index[(size_t)n_edges + e];
  float v = h[(size_t)src * DD + d];
#pragma unroll
  for (int f = 0; f < NBF; ++f) {
    int idx = edge_attr[(size_t)e * NBF + f];
    v += bond_emb_l[((size_t)f * BVOC + idx) * DD + d];
  }
  v = fmaxf(v, 0.0f);
  atomicAdd(&agg[(size_t)dst * DD + d], v);
}

// ---------------------------------------------------------------------------
// Zero the BN statistics scratch (768 floats)
// ---------------------------------------------------------------------------
__global__ void zero_stats_kernel(float* stats) {
  if (threadIdx.x < 768) stats[threadIdx.x] = 0.0f;
}

// ---------------------------------------------------------------------------
// GEMM1: Y[n,128] = Z[n,64] @ W[64,128] + bias, plus BN column sum/sumsq.
// Block = 256 thr = 8 waves; block does 16 rows x 128 cols; wave w -> cols [16w,16w+16).
// Uses V_WMMA_F32_16X16X4_F32 chained over K=64 (exact fp32 math).
// ---------------------------------------------------------------------------
__global__ __launch_bounds__(256) void gemm1_bn_kernel(
    const float* __restrict__ Z, const float* __restrict__ W,
    const float* __restrict__ bias, float* __restrict__ Y,
    float* __restrict__ colsum, float* __restrict__ colsq, int n) {
  __shared__ float As[16 * 65];  // padded stride -> conflict-free frag reads
  const int row0 = blockIdx.x * 16;
  for (int i = threadIdx.x; i < 16 * 64; i += 256) {
    int r = i >> 6, c = i & 63;
    int row = row0 + r;
    As[r * 65 + c] = (row < n) ? Z[(size_t)row * DD + c] : 0.0f;
  }
  __syncthreads();

  const int lane = threadIdx.x & 31;
  const int wave = threadIdx.x >> 5;
  const int nn = lane & 15;  // A-row / B-col / C-col within tile
  const int hi = lane >> 4;  // lane half selects K pair {0,1} vs {2,3}
  const int col = wave * 16 + nn;

  v8f acc = {0.f, 0.f, 0.f, 0.f, 0.f, 0.f, 0.f, 0.f};
#pragma unroll
  for (int kk = 0; kk < DD; kk += 4) {
    int ka = kk + 2 * hi;
    v2f a, b;
    a.x = As[nn * 65 + ka + 0];
    a.y = As[nn * 65 + ka + 1];
    b.x = W[(size_t)(ka + 0) * DD2 + col];
    b.y = W[(size_t)(ka + 1) * DD2 + col];
    acc = __builtin_amdgcn_wmma_f32_16x16x4_f32(false, a, false, b, (short)0,
                                                acc, false, false);
  }
  const float bcol = bias[col];
  float s = 0.0f, sq = 0.0f;
#pragma unroll
  for (int r = 0; r < 8; ++r) {
    int row = row0 + r + 8 * hi;
    float v = acc[r] + bcol;
    if (row < n) {
      Y[(size_t)row * DD2 + col] = v;
      s += v;
      sq += v * v;
    }
  }
  // lanes l and l+16 hold the same column, different rows: merge halves
  s += __shfl_xor(s, 16, 32);
  sq += __shfl_xor(sq, 16, 32);
  if (hi == 0) {
    atomicAdd(&colsum[col], s);
    atomicAdd(&colsq[col], sq);
  }
}

// ---------------------------------------------------------------------------
// BN finalize: scale = g*rsqrt(var+eps); shift = b - mean*scale
// ---------------------------------------------------------------------------
__global__ void bn_finalize_kernel(const float* __restrict__ colsum,
                                   const float* __restrict__ colsq,
                                   const float* __restrict__ g,
                                   const float* __restrict__ b,
                                   float* __restrict__ sc_out,
                                   float* __restrict__ sh_out, int c,
                                   float inv_n) {
  int i = threadIdx.x;
  if (i < c) {
    float mean = colsum[i] * inv_n;
    float var = colsq[i] * inv_n - mean * mean;
    float sc = g[i] * rsqrtf(var + 1e-5f);
    sc_out[i] = sc;
    sh_out[i] = b[i] - mean * sc;
  }
}

// ---------------------------------------------------------------------------
// GEMM2: A = relu(BN1(Y1)) applied while staging into LDS;
// Y2[n,64] = A[n,128] @ W[128,64] + bias, plus BN2 column stats.
// Block = 8 waves; 32 rows x 64 cols per block; wave -> (rowtile, coltile).
// ---------------------------------------------------------------------------
__global__ __launch_bounds__(256) void gemm2_bn_kernel(
    const float* __restrict__ Y1, const float* __restrict__ s1,
    const float* __restrict__ t1, const float* __restrict__ W,
    const float* __restrict__ bias, float* __restrict__ Y2,
    float* __restrict__ colsum, float* __restrict__ colsq, int n) {
  __shared__ float As[32 * 129];
  const int row0 = blockIdx.x * 32;
  for (int i = threadIdx.x; i < 32 * 128; i += 256) {
    int r = i >> 7, c = i & 127;
    int row = row0 + r;
    float v = 0.0f;
    if (row < n) v = fmaxf(Y1[(size_t)row * DD2 + c] * s1[c] + t1[c], 0.0f);
    As[r * 129 + c] = v;
  }
  __syncthreads();

  const int lane = threadIdx.x & 31;
  const int wave = threadIdx.x >> 5;
  const int nn = lane & 15;
  const int hi = lane >> 4;
  const int rt = wave >> 2;  // row tile 0..1
  const int ct = wave & 3;   // col tile 0..3
  const int col = ct * 16 + nn;
  const float* Arow = &As[(rt * 16 + nn) * 129];

  v8f acc = {0.f, 0.f, 0.f, 0.f, 0.f, 0.f, 0.f, 0.f};
#pragma unroll
  for (int kk = 0; kk < DD2; kk += 4) {
    int ka = kk + 2 * hi;
    v2f a, b;
    a.x = Arow[ka + 0];
    a.y = Arow[ka + 1];
    b.x = W[(size_t)(ka + 0) * DD + col];
    b.y = W[(size_t)(ka + 1) * DD + col];
    acc = __builtin_amdgcn_wmma_f32_16x16x4_f32(false, a, false, b, (short)0,
                                                acc, false, false);
  }
  const float bcol = bias[col];
  float s = 0.0f, sq = 0.0f;
#pragma unroll
  for (int r = 0; r < 8; ++r) {
    int row = row0 + rt * 16 + r + 8 * hi;
    float v = acc[r] + bcol;
    if (row < n) {
      Y2[(size_t)row * DD + col] = v;
      s += v;
      sq += v * v;
    }
  }
  s += __shfl_xor(s, 16, 32);
  sq += __shfl_xor(sq, 16, 32);
  if (hi == 0) {
    atomicAdd(&colsum[col], s);
    atomicAdd(&colsq[col], sq);
  }
}

// ---------------------------------------------------------------------------
// Apply outer BN (+optional relu): h = Y2 * s2[col] + t2[col]
// ---------------------------------------------------------------------------
__global__ __launch_bounds__(256) void bn_apply_kernel(
    const float* __restrict__ Y2, const float* __restrict__ s2,
    const float* __restrict__ t2, float* __restrict__ h, long long total,
    int do_relu) {
  long long i = (long long)blockIdx.x * blockDim.x + threadIdx.x;
  if (i >= total) return;
  int c = (int)(i & 63);
  float v = Y2[i] * s2[c] + t2[c];
  if (do_relu) v = fmaxf(v, 0.0f);
  h[i] = v;
}

// ---------------------------------------------------------------------------
extern "C" void kernel_launch(void* const* d_in, const int* in_sizes, int n_in,
                              void* d_out, int out_size, void* d_ws,
                              size_t ws_size, hipStream_t stream) {
  const int* x = (const int*)d_in[0];
  const int* edge_index = (const int*)d_in[1];
  const int* edge_attr = (const int*)d_in[2];
  const float* atom_emb = (const float*)d_in[3];
  const float* bond_emb = (const float*)d_in[4];
  const float* lin1_w = (const float*)d_in[5];
  const float* lin1_b = (const float*)d_in[6];
  const float* bn1_g = (const float*)d_in[7];
  const float* bn1_b = (const float*)d_in[8];
  const float* lin2_w = (const float*)d_in[9];
  const float* lin2_b = (const float*)d_in[10];
  const float* eps = (const float*)d_in[11];
  const float* bn_g = (const float*)d_in[12];
  const float* bn_b = (const float*)d_in[13];

  const int n = in_sizes[0] / NAF;        // 50000 nodes
  const int ne = in_sizes[1] / 2;         // 800000 edges
  const long long total = (long long)n * DD;

  float* ws = (float*)d_ws;
  float* agg = ws;                               // n*64
  float* y1 = agg + (size_t)n * DD;              // n*128
  float* y2 = y1 + (size_t)n * DD2;              // n*64
  float* stats = y2 + (size_t)n * DD;            // 768 floats
  // stats layout: [0,128) sum1 [128,256) sq1 [256,384) s1 [384,512) t1
  //               [512,576) sum2 [576,640) sq2 [640,704) s2 [704,768) t2
  float* h = (float*)d_out;

  const float inv_n = 1.0f / (float)n;
  const int ew_blocks = (int)((total + 255) / 256);

  atom_enc_kernel<<<(n + 3) / 4, 256, 0, stream>>>(x, atom_emb, h, n);

  for (int l = 0; l < NLAYERS; ++l) {
    zero_stats_kernel<<<1, 768, 0, stream>>>(stats);
    gin_init_kernel<<<ew_blocks, 256, 0, stream>>>(h, eps, l, agg, total);
    msg_scatter_kernel<<<(ne + 3) / 4, 256, 0, stream>>>(
        h, edge_index, edge_attr, bond_emb + (size_t)l * NBF * BVOC * DD, agg,
        ne);
    gemm1_bn_kernel<<<(n + 15) / 16, 256, 0, stream>>>(
        agg, lin1_w + (size_t)l * DD * DD2, lin1_b + (size_t)l * DD2, y1,
        stats + 0, stats + 128, n);
    bn_finalize_kernel<<<1, 128, 0, stream>>>(stats + 0, stats + 128,
                                              bn1_g + (size_t)l * DD2,
                                              bn1_b + (size_t)l * DD2,
                                              stats + 256, stats + 384, DD2,
                                              inv_n);
    gemm2_bn_kernel<<<(n + 31) / 32, 256, 0, stream>>>(
        y1, stats + 256, stats + 384, lin2_w + (size_t)l * DD2 * DD,
        lin2_b + (size_t)l * DD, y2, stats + 512, stats + 576, n);
    bn_finalize_kernel<<<1, 64, 0, stream>>>(stats + 512, stats + 576,
                                             bn_g + (size_t)l * DD,
                                             bn_b + (size_t)l * DD, stats + 640,
                                             stats + 704, DD, inv_n);
    bn_apply_kernel<<<ew_blocks, 256, 0, stream>>>(
        y2, stats + 640, stats + 704, h, total, (l < NLAYERS - 1) ? 1 : 0);
  }
}